// disc_loss_60705067761899
// MI455X (gfx1250) — compile-verified
//
#include <hip/hip_runtime.h>
#include <math.h>

#define DD 4
#define HH 512
#define NPIX (HH*HH)
#define BB 32
#define PPB 64          // partitions (blocks) per batch image
#define THREADS 256     // 8 waves of 32

typedef __attribute__((ext_vector_type(2))) float v2f;
typedef __attribute__((ext_vector_type(8))) float v8f;

__device__ __forceinline__ float wave_sum32(float v) {
#pragma unroll
    for (int off = 16; off > 0; off >>= 1) v += __shfl_down(v, off, 32);
    return v;
}

// ---------------- kernel 0: zero workspace + output ----------------
__global__ void k_zero(float* __restrict__ ws, float* __restrict__ out) {
    const int i = threadIdx.x + blockIdx.x * blockDim.x;
    if (i < BB * 32) ws[i] = 0.0f;
    if (i == 0) out[0] = 0.0f;
}

// ---------------- kernel 1: counts + segment feature sums ----------------
// ws layout per batch b (stride 32 floats):
//   [4..15]  ssum[d][f] for d=1..3 (row 0 of ssum is zeroed per reference)
//   [17..19] counts[d]  for d=1..3
//   [21..23] dsum[d]    for d=1..3 (filled by pass 2)
__global__ __launch_bounds__(THREADS)
void k_pass1(const float* __restrict__ emb, const int* __restrict__ lab,
             float* __restrict__ ws) {
    const int b    = blockIdx.x / PPB;
    const int part = blockIdx.x % PPB;
    const int chunk = NPIX / PPB;            // 4096 pixels per block
    const int base  = part * chunk;
    const float* e = emb + (size_t)b * DD * NPIX;
    const int*   L = lab + (size_t)b * NPIX;

    float cnt[3] = {0.0f, 0.0f, 0.0f};
    float s[3][4] = {};

    for (int g = threadIdx.x; g < chunk / 4; g += THREADS) {
        const int p = base + g * 4;
        const int4   sg = *(const int4*)(L + p);
        const float4 x0 = *(const float4*)(e + p);
        const float4 x1 = *(const float4*)(e + NPIX + p);
        const float4 x2 = *(const float4*)(e + 2 * NPIX + p);
        const float4 x3 = *(const float4*)(e + 3 * NPIX + p);
        __builtin_prefetch(e + p + 4 * THREADS, 0, 1);   // global_prefetch_b8

        const int   segv[4] = { sg.x, sg.y, sg.z, sg.w };
        const float f0[4] = { x0.x, x0.y, x0.z, x0.w };
        const float f1[4] = { x1.x, x1.y, x1.z, x1.w };
        const float f2[4] = { x2.x, x2.y, x2.z, x2.w };
        const float f3[4] = { x3.x, x3.y, x3.z, x3.w };
#pragma unroll
        for (int q = 0; q < 4; ++q) {
            const int seg = segv[q];
#pragma unroll
            for (int d = 1; d <= 3; ++d) {
                const float m = (seg == d) ? 1.0f : 0.0f;
                cnt[d - 1]  += m;
                s[d - 1][0] += m * f0[q];
                s[d - 1][1] += m * f1[q];
                s[d - 1][2] += m * f2[q];
                s[d - 1][3] += m * f3[q];
            }
        }
    }

    float vals[15];
#pragma unroll
    for (int d = 0; d < 3; ++d)
#pragma unroll
        for (int f = 0; f < 4; ++f) vals[d * 4 + f] = s[d][f];
    vals[12] = cnt[0]; vals[13] = cnt[1]; vals[14] = cnt[2];

#pragma unroll
    for (int k = 0; k < 15; ++k) vals[k] = wave_sum32(vals[k]);

    __shared__ float red[8][15];
    const int wid = threadIdx.x >> 5, lane = threadIdx.x & 31;
    if (lane == 0) {
#pragma unroll
        for (int k = 0; k < 15; ++k) red[wid][k] = vals[k];
    }
    __syncthreads();
    if (threadIdx.x < 15) {
        float t = 0.0f;
#pragma unroll
        for (int wv = 0; wv < 8; ++wv) t += red[wv][threadIdx.x];
        const int k = threadIdx.x;
        const int slot = (k < 12) ? (((k >> 2) + 1) * 4 + (k & 3)) : (16 + (k - 11));
        atomicAdd(&ws[b * 32 + slot], t);
    }
}

// ---------------- kernel 2: hinge-variance segment sums ----------------
__global__ __launch_bounds__(THREADS)
void k_pass2(const float* __restrict__ emb, const int* __restrict__ lab,
             float* __restrict__ ws) {
    const int b    = blockIdx.x / PPB;
    const int part = blockIdx.x % PPB;
    const int chunk = NPIX / PPB;
    const int base  = part * chunk;
    const float* e = emb + (size_t)b * DD * NPIX;
    const int*   L = lab + (size_t)b * NPIX;

    __shared__ float mu_s[3][4];
    if (threadIdx.x < 12) {
        const int d = threadIdx.x >> 2, f = threadIdx.x & 3;
        const float c = ws[b * 32 + 17 + d];
        mu_s[d][f] = ws[b * 32 + (d + 1) * 4 + f] / (c > 0.0f ? c : 1.0f);
    }
    __syncthreads();
    float mu[3][4];
#pragma unroll
    for (int d = 0; d < 3; ++d)
#pragma unroll
        for (int f = 0; f < 4; ++f) mu[d][f] = mu_s[d][f];

    float a1 = 0.0f, a2 = 0.0f, a3 = 0.0f;

    for (int g = threadIdx.x; g < chunk / 4; g += THREADS) {
        const int p = base + g * 4;
        const int4   sg = *(const int4*)(L + p);
        const float4 x0 = *(const float4*)(e + p);
        const float4 x1 = *(const float4*)(e + NPIX + p);
        const float4 x2 = *(const float4*)(e + 2 * NPIX + p);
        const float4 x3 = *(const float4*)(e + 3 * NPIX + p);
        __builtin_prefetch(e + p + 4 * THREADS, 0, 1);

        const int   segv[4] = { sg.x, sg.y, sg.z, sg.w };
        const float f0[4] = { x0.x, x0.y, x0.z, x0.w };
        const float f1[4] = { x1.x, x1.y, x1.z, x1.w };
        const float f2[4] = { x2.x, x2.y, x2.z, x2.w };
        const float f3[4] = { x3.x, x3.y, x3.z, x3.w };
#pragma unroll
        for (int q = 0; q < 4; ++q) {
            const int seg = segv[q];
            // select mu[seg] (mu row 0 is all zeros)
            const float m0 = (seg == 1) ? mu[0][0] : (seg == 2) ? mu[1][0] : (seg == 3) ? mu[2][0] : 0.0f;
            const float m1 = (seg == 1) ? mu[0][1] : (seg == 2) ? mu[1][1] : (seg == 3) ? mu[2][1] : 0.0f;
            const float m2 = (seg == 1) ? mu[0][2] : (seg == 2) ? mu[1][2] : (seg == 3) ? mu[2][2] : 0.0f;
            const float m3 = (seg == 1) ? mu[0][3] : (seg == 2) ? mu[1][3] : (seg == 3) ? mu[2][3] : 0.0f;
            const float q0 = m0 - f0[q], q1 = m1 - f1[q], q2 = m2 - f2[q], q3 = m3 - f3[q];
            const float d2 = q0 * q0 + q1 * q1 + q2 * q2 + q3 * q3;
            const float dist = sqrtf(d2);
            const float h = fmaxf(dist - 0.5f, 0.0f);   // DELTA_V = 0.5
            const float hh = h * h;
            a1 += (seg == 1) ? hh : 0.0f;
            a2 += (seg == 2) ? hh : 0.0f;
            a3 += (seg == 3) ? hh : 0.0f;
        }
    }

    float vals[3] = { a1, a2, a3 };
#pragma unroll
    for (int k = 0; k < 3; ++k) vals[k] = wave_sum32(vals[k]);

    __shared__ float red[8][3];
    const int wid = threadIdx.x >> 5, lane = threadIdx.x & 31;
    if (lane == 0) {
#pragma unroll
        for (int k = 0; k < 3; ++k) red[wid][k] = vals[k];
    }
    __syncthreads();
    if (threadIdx.x < 3) {
        float t = 0.0f;
#pragma unroll
        for (int wv = 0; wv < 8; ++wv) t += red[wv][threadIdx.x];
        atomicAdd(&ws[b * 32 + 21 + threadIdx.x], t);
    }
}

// ---------------- kernel 3: finalize (one wave per batch item) ----------------
// Pairwise mu-distance Gram = Diff(16x4) x Diff^T(4x16) via V_WMMA_F32_16X16X4_F32.
// 32-bit A layout: lanes 0-15 hold K=0 (v0), K=1 (v1); lanes 16-31 hold K=2,3.
// For B = Diff^T (KxN) the mirrored layout makes the B operand identical to A.
__global__ __launch_bounds__(1024)
void k_final(const float* __restrict__ ws, float* __restrict__ out) {
    const int tid  = threadIdx.x;
    const int w    = tid >> 5;        // batch index (32 waves)
    const int lane = tid & 31;

    __shared__ float gram[BB][256];
    __shared__ float mu_sh[BB][16];
    __shared__ float lossb[BB];

    const float* wsb = ws + w * 32;
    if (lane < 16) {
        const int d = lane >> 2, f = lane & 3;
        float m = 0.0f;
        if (d > 0) {
            const float c = wsb[16 + d];
            m = wsb[d * 4 + f] / (c > 0.0f ? c : 1.0f);
        }
        mu_sh[w][lane] = m;
    }
    __syncthreads();

    // masked Diff row r = i*4+j : diff[c] = (mu_j[c]-mu_i[c]) iff both comps != 0
    const int r  = lane & 15;
    const int i  = r >> 2, j = r & 3;
    const int cb = (lane < 16) ? 0 : 2;
    const float mi0 = mu_sh[w][i * 4 + cb],     mj0 = mu_sh[w][j * 4 + cb];
    const float mi1 = mu_sh[w][i * 4 + cb + 1], mj1 = mu_sh[w][j * 4 + cb + 1];
    v2f A;
    A.x = (mi0 != 0.0f && mj0 != 0.0f) ? (mj0 - mi0) : 0.0f;
    A.y = (mi1 != 0.0f && mj1 != 0.0f) ? (mj1 - mi1) : 0.0f;

    v8f C = {0.0f, 0.0f, 0.0f, 0.0f, 0.0f, 0.0f, 0.0f, 0.0f};
    // EXEC is all ones here (no divergence); exact f32 16x16x4 matrix op.
    C = __builtin_amdgcn_wmma_f32_16x16x4_f32(false, A, false, A,
                                              (short)0, C, false, false);
#pragma unroll
    for (int v = 0; v < 8; ++v) {
        const int row = (lane < 16) ? v : (v + 8);
        gram[w][row * 16 + (lane & 15)] = C[v];
    }
    __syncthreads();

    if (lane == 0) {
        float mnsum = 0.0f, msum = 0.0f;
        for (int rr = 0; rr < 16; ++rr) {
            const int ii = rr >> 2, jj = rr & 3;
            float asum = 0.0f;
            for (int c = 0; c < 4; ++c) {
                const float a  = mu_sh[w][ii * 4 + c];
                const float bb = mu_sh[w][jj * 4 + c];
                const float dc = (a != 0.0f && bb != 0.0f) ? (bb - a) : 0.0f;
                asum += fabsf(dc);
            }
            if (asum != 0.0f) {
                const float d2  = gram[w][rr * 16 + rr];        // ||diff_r||^2
                const float nrm = (d2 > 0.0f) ? sqrtf(d2) : 0.0f;
                float mn = fmaxf(6.0f - nrm, 0.0f);             // 2*DELTA_D = 6
                mnsum += mn * mn;
                msum  += 1.0f;
            }
        }
        const float l_dist = (msum > 0.0f) ? (mnsum / fmaxf(msum, 1.0f)) : 0.0f;

        float npres = 0.0f, lv = 0.0f;
        for (int d = 1; d <= 3; ++d) {
            const float c = wsb[16 + d];
            if (c > 0.0f) { npres += 1.0f; lv += wsb[20 + d] / c; }
        }
        const float l_var = lv / fmaxf(npres, 1.0f);

        float reg = 0.0f;
        for (int d = 0; d < 4; ++d) {
            float sq = 0.0f;
            for (int c = 0; c < 4; ++c) { const float m = mu_sh[w][d * 4 + c]; sq += m * m; }
            reg += (sq > 0.0f) ? sqrtf(sq) : 0.0f;
        }
        reg *= 0.25f;

        const float loss = l_var + l_dist + 0.001f * reg;   // PARAM_VAR=PARAM_DIST=1
        lossb[w] = (npres > 0.0f) ? loss : 0.0f;
    }
    __syncthreads();

    if (tid == 0) {
        float t = 0.0f;
#pragma unroll
        for (int k = 0; k < BB; ++k) t += lossb[k];
        out[0] = t / (float)BB;     // deterministic ordered batch mean
    }
}

extern "C" void kernel_launch(void* const* d_in, const int* in_sizes, int n_in,
                              void* d_out, int out_size, void* d_ws, size_t ws_size,
                              hipStream_t stream) {
    const float* emb = (const float*)d_in[0];   // pix_embedding [B,4,512,512] f32
    // d_in[1] = y (unused by the reference)
    const int*   lab = (const int*)d_in[2];     // instance_label [B,512,512] int
    float* out = (float*)d_out;
    float* ws  = (float*)d_ws;                  // 32 floats per batch = 4 KB

    k_zero <<<1, 1024, 0, stream>>>(ws, out);
    k_pass1<<<BB * PPB, THREADS, 0, stream>>>(emb, lab, ws);
    k_pass2<<<BB * PPB, THREADS, 0, stream>>>(emb, lab, ws);
    k_final<<<1, 1024, 0, stream>>>(ws, out);
}